// SiameseModel_38345468019271
// MI455X (gfx1250) — compile-verified
//
#include <hip/hip_runtime.h>

typedef float v2f __attribute__((ext_vector_type(2)));
typedef float v8f __attribute__((ext_vector_type(8)));

#define BB 384
#define DD 128
#define NT (BB / 16)     // 24 tiles per dim
#define MARGIN 0.5f

// ---------------------------------------------------------------------------
// Kernel 1: G = E * E^T using V_WMMA_F32_16X16X4_F32 (full fp32 precision).
// One wave32 per 16x16 output tile; K-loop of 32 chained WMMAs.
// A-layout (16x4 f32): lane&15 = M row; lanes 0-15 hold K={0,1} in v[0],v[1];
// lanes 16-31 hold K={2,3}. B (4x16) mirrors with lane&15 = N column.
// C/D (16x16 f32, 8 VGPRs): VGPR r holds M = r + 8*(lane>>4), N = lane&15.
// ---------------------------------------------------------------------------
__global__ __launch_bounds__(32) void gram_wmma(const float* __restrict__ E,
                                                float* __restrict__ G) {
    const int tile = blockIdx.x;
    const int m0 = (tile / NT) * 16;
    const int n0 = (tile % NT) * 16;
    const int lane = threadIdx.x;     // 0..31, full wave active (EXEC all 1s)
    const int l    = lane & 15;
    const int half = lane >> 4;

    const float* arow = E + (m0 + l) * DD;   // A tile row (M = m0+l)
    const float* brow = E + (n0 + l) * DD;   // B tile "column" = row of E

    v8f c = {};
#pragma unroll 4
    for (int k = 0; k < DD; k += 4) {
        const int kb = k + 2 * half;
        v2f av = *(const v2f*)(arow + kb);   // aligned 8B load
        v2f bv = *(const v2f*)(brow + kb);
        // 8 args: (neg_a, A, neg_b, B, c_mod, C, reuse_a, reuse_b)
        c = __builtin_amdgcn_wmma_f32_16x16x4_f32(
                false, av, false, bv, (short)0, c, false, false);
    }

    float* gout = G + (m0 + 8 * half) * BB + n0 + l;
#pragma unroll
    for (int r = 0; r < 8; ++r) gout[r * BB] = c[r];
}

// ---------------------------------------------------------------------------
// Kernel 2: sq = diag(G); zero the global accumulators (must happen every
// call: harness poisons ws once and replays the graph many times).
// ---------------------------------------------------------------------------
__global__ void diag_init(const float* __restrict__ G,
                          float* __restrict__ sq, float* __restrict__ acc) {
    const int t = threadIdx.x;
    if (t < BB) sq[t] = G[t * (BB + 1)];
    if (t < 2) acc[t] = 0.0f;
}

// ---------------------------------------------------------------------------
// Kernel 3: one block per anchor a. d(a,x) built on the fly in LDS from the
// Gram row; positives list (~5 entries) built once; each thread sweeps
// negatives. Valid mask reduces to {p != a, lab[p]==lab[a], lab[n]!=lab[a]}
// since the label conditions already imply a!=n and p!=n.
// ---------------------------------------------------------------------------
__global__ __launch_bounds__(256) void triplet_reduce(
        const float* __restrict__ G, const float* __restrict__ sq,
        const int* __restrict__ labels, float* __restrict__ acc) {
    __shared__ float s_d[BB];
    __shared__ int   s_lab[BB];
    __shared__ int   s_pos[BB];
    __shared__ int   s_npos;
    __shared__ float s_sum[256];
    __shared__ float s_cnt[256];

    const int a   = blockIdx.x;
    const int tid = threadIdx.x;
    const float sqa = sq[a];
    const float* grow = G + a * BB;

    for (int x = tid; x < BB; x += 256) {
        s_lab[x] = labels[x];
        float dd = sqa - 2.0f * grow[x] + sq[x];
        s_d[x] = dd > 0.0f ? dd : 0.0f;
    }
    __syncthreads();

    const int la = s_lab[a];
    if (tid == 0) {
        int np = 0;
        for (int p = 0; p < BB; ++p)
            if (p != a && s_lab[p] == la) s_pos[np++] = p;
        s_npos = np;
    }
    __syncthreads();

    const int np = s_npos;
    float lsum = 0.0f, lcnt = 0.0f;
    for (int n = tid; n < BB; n += 256) {
        if (s_lab[n] == la) continue;          // excludes n==a too
        const float dan = s_d[n];
        for (int i = 0; i < np; ++i) {
            float t = s_d[s_pos[i]] - dan + MARGIN;
            float v = t > 0.0f ? t : 0.0f;
            lsum += v;
            if (v > 1e-16f) lcnt += 1.0f;
        }
    }

    s_sum[tid] = lsum;
    s_cnt[tid] = lcnt;
    __syncthreads();
    for (int s = 128; s > 0; s >>= 1) {
        if (tid < s) {
            s_sum[tid] += s_sum[tid + s];
            s_cnt[tid] += s_cnt[tid + s];
        }
        __syncthreads();
    }
    if (tid == 0) {
        atomicAdd(&acc[0], s_sum[0]);
        atomicAdd(&acc[1], s_cnt[0]);
    }
}

// ---------------------------------------------------------------------------
// Kernel 4: loss = sum / (num_positive + 1e-16)
// ---------------------------------------------------------------------------
__global__ void finalize(const float* __restrict__ acc, float* __restrict__ out) {
    out[0] = acc[0] / (acc[1] + 1e-16f);
}

extern "C" void kernel_launch(void* const* d_in, const int* in_sizes, int n_in,
                              void* d_out, int out_size, void* d_ws, size_t ws_size,
                              hipStream_t stream) {
    (void)in_sizes; (void)n_in; (void)out_size; (void)ws_size;
    const float* E      = (const float*)d_in[0];
    const int*   labels = (const int*)d_in[1];
    float*       out    = (float*)d_out;

    // Workspace layout: G [384*384] | sq [384] | acc [2]
    float* G   = (float*)d_ws;
    float* sq  = G + BB * BB;
    float* acc = sq + BB;

    gram_wmma<<<NT * NT, 32, 0, stream>>>(E, G);
    diag_init<<<1, BB, 0, stream>>>(G, sq, acc);
    triplet_reduce<<<BB, 256, 0, stream>>>(G, sq, labels, acc);
    finalize<<<1, 1, 0, stream>>>(acc, out);
}